// SAGE_43868795961414
// MI455X (gfx1250) — compile-verified
//
#include <hip/hip_runtime.h>
#include <hip/hip_bf16.h>

// ---------------------------------------------------------------------------
// SAGE GNN (2 GCNs x [SAGEConv->BN->ReLU x2, SAGEConv 128->40]) for gfx1250.
// fp32 matrix path: V_WMMA_F32_16X16X4_F32 (exact precision match with the
// fp32 reference; GEMMs are ~16 GF total and NOT the bottleneck -- the
// scatter-mean passes are, and the ~90MB working set is L2-resident).
// H-tile staging uses GLOBAL_LOAD_ASYNC_TO_LDS_B128 (ASYNCcnt) when the
// toolchain exposes it; A-tile staging fuses the 1/deg scaling.
// ---------------------------------------------------------------------------

typedef float v2f __attribute__((ext_vector_type(2)));
typedef float v8f __attribute__((ext_vector_type(8)));

#define NNODES 50000
#define CH     128
#define OUTCH  40
#define EPSBN  1e-5f
#define LDSPITCH 132   // 128 + 4 floats: kills stride-128 bank conflicts; rows stay 16B-aligned

#if defined(__HIP_DEVICE_COMPILE__) && \
    __has_builtin(__builtin_amdgcn_global_load_async_to_lds_b128) && \
    __has_builtin(__builtin_amdgcn_s_wait_asynccnt)
#define HAVE_ASYNC_LDS 1
// parameter types per clang diagnostic: (v4i AS1*, v4i AS3*, imm int, imm int)
typedef int v4i_b __attribute__((vector_size(16)));
typedef __attribute__((address_space(1))) v4i_b* g_v4i_ptr;
typedef __attribute__((address_space(3))) v4i_b* l_v4i_ptr;
#else
#define HAVE_ASYNC_LDS 0
#endif

// ---------------------------------------------------------------- degree ---
__global__ __launch_bounds__(256) void deg_kernel(const int* __restrict__ dst,
                                                  float* __restrict__ deg, int E) {
    int e = blockIdx.x * blockDim.x + threadIdx.x;
    if (e < E) atomicAdd(&deg[dst[e]], 1.0f);
}

// ------------------------------------------------------------ scatter-add ---
// one wave32 per edge, each lane moves 4 contiguous floats (128 ch total)
__global__ __launch_bounds__(256) void scatter_kernel(const float* __restrict__ h,
                                                      const int* __restrict__ src,
                                                      const int* __restrict__ dst,
                                                      float* __restrict__ agg, int E) {
    long long gid = (long long)blockIdx.x * blockDim.x + threadIdx.x;
    int edge = (int)(gid >> 5);
    int lane = (int)(gid & 31);
    if (edge >= E) return;
    int s = src[edge];
    int d = dst[edge];
    const float4 v = *(const float4*)(h + (size_t)s * CH + lane * 4);
    float* o = agg + (size_t)d * CH + lane * 4;
    atomicAdd(o + 0, v.x);
    atomicAdd(o + 1, v.y);
    atomicAdd(o + 2, v.z);
    atomicAdd(o + 3, v.w);
}

// -------------------------------------------- LDS staging (16 x 128 tile) ---
// H-tile: raw copy -> async global->LDS DMA path when available (ASYNCcnt).
__device__ __forceinline__ void stage_h_tile(const float* __restrict__ g,
                                             float* lds, int rowBase, int nRows) {
    for (int i = threadIdx.x; i < 16 * 32; i += 256) {   // 16B chunks
        int r = i >> 5, c4 = (i & 31) * 4;
        int row = rowBase + r;
        float* dst = lds + r * LDSPITCH + c4;
        if (row < nRows) {
#if HAVE_ASYNC_LDS
            __builtin_amdgcn_global_load_async_to_lds_b128(
                (g_v4i_ptr)(g + (size_t)row * CH + c4),
                (l_v4i_ptr)dst, 0, 0);
#else
            *(float4*)dst = *(const float4*)(g + (size_t)row * CH + c4);
#endif
        } else {
            *(float4*)dst = float4{0.f, 0.f, 0.f, 0.f};
        }
    }
}

// A-tile: mean aggregation = running sums scaled by 1/max(deg,1) on the way in.
__device__ __forceinline__ void stage_a_tile(const float* __restrict__ agg,
                                             const float* __restrict__ deg,
                                             float* lds, int rowBase, int nRows) {
    for (int i = threadIdx.x; i < 16 * CH; i += 256) {
        int r = i >> 7, c = i & (CH - 1);
        int row = rowBase + r;
        float av = 0.f;
        if (row < nRows) {
            float inv = 1.0f / fmaxf(deg[row], 1.0f);
            av = agg[(size_t)row * CH + c] * inv;
        }
        lds[r * LDSPITCH + c] = av;
    }
}

__device__ __forceinline__ void stage_fence() {
#if HAVE_ASYNC_LDS
    __builtin_amdgcn_s_wait_asynccnt(0);
#endif
    __syncthreads();
}

// ------------------------------------------------- WMMA GEMM (128 -> 128) ---
// z[16 rows x 128 cols] = (agg/deg) @ Wl + h @ Wr + b
// block = 256 threads = 8 wave32; wave w owns output N-tile [16w, 16w+16)
__global__ __launch_bounds__(256) void gemm128_kernel(const float* __restrict__ hin,
                                                      const float* __restrict__ agg,
                                                      const float* __restrict__ deg,
                                                      const float* __restrict__ Wl,
                                                      const float* __restrict__ Wr,
                                                      const float* __restrict__ bias,
                                                      float* __restrict__ z, int nRows) {
    __shared__ float As[16 * LDSPITCH];
    __shared__ float Hs[16 * LDSPITCH];
    const int rowBase = blockIdx.x * 16;

    stage_h_tile(hin, Hs, rowBase, nRows);
    stage_a_tile(agg, deg, As, rowBase, nRows);
    stage_fence();

    const int wave  = threadIdx.x >> 5;          // wave32
    const int lane  = threadIdx.x & 31;
    const int nBase = wave * 16;
    const int n     = lane & 15;                 // B/C/D column
    const int m     = lane & 15;                 // A row
    const int khalf = (lane >> 4) * 2;           // 16x4 f32 A layout: K msb from lane half

    v8f acc;
    const float bv = bias[nBase + n];
#pragma unroll
    for (int r = 0; r < 8; ++r) acc[r] = bv;

#pragma unroll 4
    for (int k = 0; k < CH; k += 4) {
        const int kb = k + khalf;
        v2f a, b;
        a[0] = As[m * LDSPITCH + kb];
        a[1] = As[m * LDSPITCH + kb + 1];
        b[0] = Wl[(size_t)kb * CH + nBase + n];
        b[1] = Wl[(size_t)(kb + 1) * CH + nBase + n];
        acc = __builtin_amdgcn_wmma_f32_16x16x4_f32(false, a, false, b,
                                                    (short)0, acc, false, false);
        a[0] = Hs[m * LDSPITCH + kb];
        a[1] = Hs[m * LDSPITCH + kb + 1];
        b[0] = Wr[(size_t)kb * CH + nBase + n];
        b[1] = Wr[(size_t)(kb + 1) * CH + nBase + n];
        acc = __builtin_amdgcn_wmma_f32_16x16x4_f32(false, a, false, b,
                                                    (short)0, acc, false, false);
    }

    const int mh = (lane >> 4) * 8;              // C/D: VGPR r -> M = r (+8 for hi lanes)
    if (rowBase + 16 <= nRows) {                 // full tile: branchless coalesced stores
#pragma unroll
        for (int r = 0; r < 8; ++r)
            z[(size_t)(rowBase + r + mh) * CH + nBase + n] = acc[r];
    } else {
#pragma unroll
        for (int r = 0; r < 8; ++r) {
            int row = rowBase + r + mh;
            if (row < nRows) z[(size_t)row * CH + nBase + n] = acc[r];
        }
    }
}

// ------------------------------------------------- WMMA GEMM (128 -> 40) ----
// 3 N-tiles (48 cols computed, cols >= 40 zero-padded / not stored)
__global__ __launch_bounds__(256) void gemmout_kernel(const float* __restrict__ hin,
                                                      const float* __restrict__ agg,
                                                      const float* __restrict__ deg,
                                                      const float* __restrict__ Wl,   // [128,40]
                                                      const float* __restrict__ Wr,   // [128,40]
                                                      const float* __restrict__ bias, // [40]
                                                      float* __restrict__ out, int nRows) {
    __shared__ float As[16 * LDSPITCH];
    __shared__ float Hs[16 * LDSPITCH];
    const int rowBase = blockIdx.x * 16;

    stage_h_tile(hin, Hs, rowBase, nRows);
    stage_a_tile(agg, deg, As, rowBase, nRows);
    stage_fence();

    const int wave = threadIdx.x >> 5;
    if (wave >= 3) return;                       // uniform wave-level exit
    const int lane  = threadIdx.x & 31;
    const int nBase = wave * 16;
    const int n     = lane & 15;
    const int col   = nBase + n;
    const int m     = lane & 15;
    const int khalf = (lane >> 4) * 2;

    v8f acc;
    const float bv = (col < OUTCH) ? bias[col] : 0.0f;
#pragma unroll
    for (int r = 0; r < 8; ++r) acc[r] = bv;

#pragma unroll 4
    for (int k = 0; k < CH; k += 4) {
        const int kb = k + khalf;
        v2f a, b;
        a[0] = As[m * LDSPITCH + kb];
        a[1] = As[m * LDSPITCH + kb + 1];
        b[0] = (col < OUTCH) ? Wl[(size_t)kb * OUTCH + col] : 0.0f;
        b[1] = (col < OUTCH) ? Wl[(size_t)(kb + 1) * OUTCH + col] : 0.0f;
        acc = __builtin_amdgcn_wmma_f32_16x16x4_f32(false, a, false, b,
                                                    (short)0, acc, false, false);
        a[0] = Hs[m * LDSPITCH + kb];
        a[1] = Hs[m * LDSPITCH + kb + 1];
        b[0] = (col < OUTCH) ? Wr[(size_t)kb * OUTCH + col] : 0.0f;
        b[1] = (col < OUTCH) ? Wr[(size_t)(kb + 1) * OUTCH + col] : 0.0f;
        acc = __builtin_amdgcn_wmma_f32_16x16x4_f32(false, a, false, b,
                                                    (short)0, acc, false, false);
    }

    const int mh = (lane >> 4) * 8;
    if (rowBase + 16 <= nRows) {
#pragma unroll
        for (int r = 0; r < 8; ++r)
            if (col < OUTCH)
                out[(size_t)(rowBase + r + mh) * OUTCH + col] = acc[r];
    } else {
#pragma unroll
        for (int r = 0; r < 8; ++r) {
            int row = rowBase + r + mh;
            if (row < nRows && col < OUTCH)
                out[(size_t)row * OUTCH + col] = acc[r];
        }
    }
}

// ----------------------------------------------------------- BN statistics --
__global__ __launch_bounds__(128) void bnstats_kernel(const float* __restrict__ z,
                                                      float* __restrict__ sums,
                                                      float* __restrict__ sumsq, int nRows) {
    const int c = threadIdx.x;                   // one thread per channel
    float s = 0.f, s2 = 0.f;
    for (int row = blockIdx.x; row < nRows; row += gridDim.x) {
        float v = z[(size_t)row * CH + c];
        s += v;
        s2 += v * v;
    }
    atomicAdd(&sums[c], s);
    atomicAdd(&sumsq[c], s2);
}

// ----------------------------------------------------- BN apply + ReLU ------
__global__ __launch_bounds__(256) void bnapply_kernel(const float* __restrict__ z,
                                                      const float* __restrict__ sums,
                                                      const float* __restrict__ sumsq,
                                                      const float* __restrict__ gamma,
                                                      const float* __restrict__ beta,
                                                      float* __restrict__ hout, int nRows) {
    long long i = (long long)blockIdx.x * blockDim.x + threadIdx.x;
    if (i >= (long long)nRows * CH) return;
    int c = (int)(i & (CH - 1));
    const float invN = 1.0f / (float)nRows;
    float mu  = sums[c] * invN;
    float var = sumsq[c] * invN - mu * mu;
    float v = (z[i] - mu) * __frsqrt_rn(var + EPSBN) * gamma[c] + beta[c];
    hout[i] = fmaxf(v, 0.0f);
}

// ---------------------------------------------------------------------------
extern "C" void kernel_launch(void* const* d_in, const int* in_sizes, int n_in,
                              void* d_out, int out_size, void* d_ws, size_t ws_size,
                              hipStream_t stream) {
    const float* x     = (const float*)d_in[0];
    const int*   edges = (const int*)d_in[1];      // [2, E]
    const float* Wl_h  = (const float*)d_in[2];    // [2,2,128,128]
    const float* Wr_h  = (const float*)d_in[3];
    const float* b_h   = (const float*)d_in[4];    // [2,2,128]
    const float* Wl_o  = (const float*)d_in[5];    // [2,128,40]
    const float* Wr_o  = (const float*)d_in[6];
    const float* b_o   = (const float*)d_in[7];    // [2,40]
    const float* gamma = (const float*)d_in[8];    // [2,2,128]
    const float* beta  = (const float*)d_in[9];
    float* out = (float*)d_out;                    // [2, N, 40]

    const int Nn = in_sizes[0] / CH;
    const int E  = in_sizes[1] / 2;
    const int* srcp = edges;
    const int* dstp = edges + E;

    // workspace carve-up (aligned to 256B)
    char* ws = (char*)d_ws;
    size_t off = 0;
    auto carve = [&](size_t bytes) {
        void* p = ws + off;
        off = (off + bytes + 255) & ~(size_t)255;
        return p;
    };
    float* deg   = (float*)carve((size_t)Nn * 4);
    float* agg   = (float*)carve((size_t)Nn * CH * 4);
    float* zbuf  = (float*)carve((size_t)Nn * CH * 4);
    float* hbuf  = (float*)carve((size_t)Nn * CH * 4);
    float* sums  = (float*)carve(CH * 4);
    float* sumsq = (float*)carve(CH * 4);
    (void)ws_size;

    const int degBlocks  = (E + 255) / 256;
    const long long scT  = (long long)E * 32;
    const int scBlocks   = (int)((scT + 255) / 256);
    const int gemmBlocks = (Nn + 15) / 16;
    const int bnBlocks   = 512;
    const int eltBlocks  = (int)(((long long)Nn * CH + 255) / 256);

    // degree (shared by every layer)
    (void)hipMemsetAsync(deg, 0, (size_t)Nn * 4, stream);
    deg_kernel<<<degBlocks, 256, 0, stream>>>(dstp, deg, E);

    for (int g = 0; g < 2; ++g) {
        const float* hcur = x;
        for (int l = 0; l < 2; ++l) {
            const int li = g * 2 + l;
            (void)hipMemsetAsync(agg, 0, (size_t)Nn * CH * 4, stream);
            scatter_kernel<<<scBlocks, 256, 0, stream>>>(hcur, srcp, dstp, agg, E);
            gemm128_kernel<<<gemmBlocks, 256, 0, stream>>>(
                hcur, agg, deg,
                Wl_h + (size_t)li * CH * CH,
                Wr_h + (size_t)li * CH * CH,
                b_h + (size_t)li * CH,
                zbuf, Nn);
            (void)hipMemsetAsync(sums, 0, CH * 4, stream);
            (void)hipMemsetAsync(sumsq, 0, CH * 4, stream);
            bnstats_kernel<<<bnBlocks, 128, 0, stream>>>(zbuf, sums, sumsq, Nn);
            bnapply_kernel<<<eltBlocks, 256, 0, stream>>>(
                zbuf, sums, sumsq,
                gamma + (size_t)li * CH,
                beta + (size_t)li * CH,
                hbuf, Nn);
            hcur = hbuf;
        }
        (void)hipMemsetAsync(agg, 0, (size_t)Nn * CH * 4, stream);
        scatter_kernel<<<scBlocks, 256, 0, stream>>>(hcur, srcp, dstp, agg, E);
        gemmout_kernel<<<gemmBlocks, 256, 0, stream>>>(
            hcur, agg, deg,
            Wl_o + (size_t)g * CH * OUTCH,
            Wr_o + (size_t)g * CH * OUTCH,
            b_o + (size_t)g * OUTCH,
            out + (size_t)g * Nn * OUTCH, Nn);
    }
}